// Spline_14087492731389
// MI455X (gfx1250) — compile-verified
//
#include <hip/hip_runtime.h>

typedef float v2f __attribute__((ext_vector_type(2)));
typedef float v8f __attribute__((ext_vector_type(8)));

#define QTOTAL (2048 * 2048)
#define EPS_TPS 1e-10f
#define HALF_LN2 0.34657359027997264f  // 0.5*ln2: phi = r2 * (0.5*ln2) * log2(r2)

// Opaque clamp: exactly one v_max_num_f32, no canonicalize (compiler can't
// prove the WMMA result non-sNaN, so fmaxf/fmed3 paths emit 2 ops/element).
__device__ __forceinline__ float clamp_lo(float x, float lo) {
  float r;
  asm("v_max_num_f32 %0, %1, %2" : "=v"(r) : "v"(x), "v"(lo));
  return r;
}

__global__ __launch_bounds__(256) void spline_tps_wmma(
    const float* __restrict__ train,  // (64,2)
    const float* __restrict__ ww,     // (64,)
    const float* __restrict__ vw,     // (3,)
    const float* __restrict__ qp,     // (Q,2)
    float* __restrict__ out)          // (Q,)
{
  const int lane = threadIdx.x & 31;
  const int wid  = (blockIdx.x * blockDim.x + threadIdx.x) >> 5;
  const int nW   = (gridDim.x * blockDim.x) >> 5;
  const int l15  = lane & 15;
  const int h    = lane >> 4;   // half-wave: carries K=2,3 and ctrl M=8..15
  const bool hi  = (h != 0);

  // --- Hoisted setup.
  // A operand (M = ctrl): row m = [-2tx, -2ty, 1, |t|^2].
  // A 16x4 layout: lanes 0-15 -> M=lane, VGPR0=K0, VGPR1=K1;
  //                lanes 16-31 -> M=lane-16, VGPR0=K2, VGPR1=K3.
  v2f a[4];
  // D element (v, lane) is ctrl index 16t + 8h + v; fold 0.5*ln2 into w.
  float pw[4][8];
#pragma unroll
  for (int t = 0; t < 4; ++t) {
    const int c = t * 16 + l15;
    const float tx = train[2 * c + 0];
    const float ty = train[2 * c + 1];
    const float tsq = tx * tx + ty * ty;
    a[t].x = hi ? 1.0f : (-2.0f * tx);
    a[t].y = hi ? tsq  : (-2.0f * ty);
#pragma unroll
    for (int v = 0; v < 8; ++v)
      pw[t][v] = HALF_LN2 * ww[t * 16 + 8 * h + v];
  }
  const float vw0 = vw[0], vw1 = vw[1], vw2 = vw[2];
  const v8f cz = {};
  float epsv = EPS_TPS;  // keep epsilon in a VGPR for the inline-asm clamp
  asm("" : "+v"(epsv));

  const int nChunks = QTOTAL / 32;  // 32 queries per wave-iteration (2 N-tiles)
  for (int ch = wid; ch < nChunks; ch += nW) {
    const int base = ch * 32;

    // Prefetch next chunk's queries (speculative; OOB silently dropped).
    __builtin_prefetch(qp + 2 * (base + nW * 32) + 2 * lane, 0, 0);

    // B operand (N = query): col n = [qx, qy, |q|^2, 1].
    // B 4x16 layout: lanes 0-15 -> N=lane, VGPR0=K0=qx, VGPR1=K1=qy;
    //                lanes 16-31 -> N=lane-16, VGPR0=K2=|q|^2, VGPR1=K3=1.
    const float2 qa = *(const float2*)(qp + 2 * (base + l15));       // tile A
    const float2 qb = *(const float2*)(qp + 2 * (base + 16 + l15));  // tile B
    v2f bA, bB;
    bA.x = hi ? (qa.x * qa.x + qa.y * qa.y) : qa.x;
    bA.y = hi ? 1.0f : qa.y;
    bB.x = hi ? (qb.x * qb.x + qb.y * qb.y) : qb.x;
    bB.y = hi ? 1.0f : qb.y;

    v8f accA = {}, accB = {};
#pragma unroll
    for (int t = 0; t < 4; ++t) {
      // D[M = v + 8h (ctrl), N = lane%16 (query)] = r2
      v8f dA = __builtin_amdgcn_wmma_f32_16x16x4_f32(
          false, a[t], false, bA, (short)0, cz, false, false);
      v8f dB = __builtin_amdgcn_wmma_f32_16x16x4_f32(
          false, a[t], false, bB, (short)0, cz, false, false);
#pragma unroll
      for (int v = 0; v < 8; ++v) {
        const float ra = dA[v];
        const float rb = dB[v];
        // phi*w = (r2 * 0.5*ln2*w) * log2(max(r2, eps)):
        //   v_max_num + v_log + v_mul + v_fmac  (4 VALU, 1 trans)
        const float la = __builtin_amdgcn_logf(clamp_lo(ra, epsv));
        const float lb = __builtin_amdgcn_logf(clamp_lo(rb, epsv));
        accA[v] = __builtin_fmaf(ra * pw[t][v], la, accA[v]);
        accB[v] = __builtin_fmaf(rb * pw[t][v], lb, accB[v]);
      }
    }

    // Reduce over ctrl: tree over the 8 VGPRs (ctrl M=8h..8h+7 across tiles)...
    float sA = ((accA[0] + accA[1]) + (accA[2] + accA[3])) +
               ((accA[4] + accA[5]) + (accA[6] + accA[7]));
    float sB = ((accB[0] + accB[1]) + (accB[2] + accB[3])) +
               ((accB[4] + accB[5]) + (accB[6] + accB[7]));
    // ...then combine the two half-waves (other 32 ctrl pts, same query).
    sA += __shfl_xor(sA, 16, 32);
    sB += __shfl_xor(sB, 16, 32);

    // Lane l's own query is (base + l): qa for lanes 0-15, qb for 16-31.
    const float qx  = hi ? qb.x : qa.x;
    const float qy  = hi ? qb.y : qa.y;
    const float rbf = hi ? sB : sA;

    out[base + lane] = rbf + vw0 * qx + vw1 * qy + vw2;  // 128B coalesced store
  }
}

extern "C" void kernel_launch(void* const* d_in, const int* in_sizes, int n_in,
                              void* d_out, int out_size, void* d_ws, size_t ws_size,
                              hipStream_t stream) {
  const float* train = (const float*)d_in[0];  // (1,64,2) f32
  const float* ww    = (const float*)d_in[1];  // (1,64,1) f32
  const float* vw    = (const float*)d_in[2];  // (1,3,1)  f32
  const float* qp    = (const float*)d_in[3];  // (1,Q,2)  f32
  float* out         = (float*)d_out;          // (1,H,W,1) f32

  // Persistent waves: 2048 blocks x 256 threads = 16384 waves, 8 chunks each;
  // A-operand / weight setup amortized across the chunk loop.
  dim3 grid(2048), block(256);
  hipLaunchKernelGGL(spline_tps_wmma, grid, block, 0, stream,
                     train, ww, vw, qp, out);
}